// LiftSplatShoot_23046794510577
// MI455X (gfx1250) — compile-verified
//
#include <hip/hip_runtime.h>
#include <hip/hip_bf16.h>

// ---------------- LiftSplatShoot voxel-pool (scatter-add) for gfx1250 ----------------
// acc layouts:
//   FINAL_LAYOUT==0 : acc[(vox)*32 + c]      (scratch, channel-contiguous atomics)
//   FINAL_LAYOUT==1 : acc[c*40000 + vox]     (final output layout, direct-atomics fallback)

#define LSS_CHUNK 192          // points per staged chunk (divides 2,770,944)
#define LSS_NX    200
#define LSS_NY    200
#define LSS_C     32
#define LSS_VOXN  (LSS_NX * LSS_NY)            // 40000
#define LSS_ACCN  ((size_t)LSS_VOXN * LSS_C)   // 1,280,000 floats (5.12 MB)

// ---- guaranteed-native non-returning f32 atomic add (L2-resident accumulator) ----
// saddr form: mem = SGPR64 base + VGPR 32-bit byte offset (acc spans only 5.12 MB).
__device__ __forceinline__ void lss_atomic_add_f32(float* __restrict__ base,
                                                   unsigned byte_off, float v) {
  asm volatile("global_atomic_add_f32 %0, %1, %2"
               :: "v"(byte_off), "v"(v), "s"(base) : "memory");
}

// ---- point binning + atomic scatter (matches reference truncation semantics) ----
template <int FINAL_LAYOUT>
__device__ __forceinline__ void lss_scatter_point(float g0, float g1, float g2,
                                                  float v, int lane,
                                                  float* __restrict__ acc) {
  // g = trunc((geom - (bx - dx/2)) / dx); lo = (-50,-50,-10); dx = (0.5,0.5,20)
  int ix = (int)((g0 + 50.0f) * 2.0f);     // /0.5 == *2 (exact)
  int iy = (int)((g1 + 50.0f) * 2.0f);
  int iz = (int)((g2 + 10.0f) / 20.0f);
  if ((unsigned)ix < (unsigned)LSS_NX && (unsigned)iy < (unsigned)LSS_NY && iz == 0) {
    unsigned vox = (unsigned)(ix * LSS_NY + iy);
    unsigned elem = FINAL_LAYOUT ? ((unsigned)lane * LSS_VOXN + vox)
                                 : (vox * LSS_C + (unsigned)lane);
    lss_atomic_add_f32(acc, elem * 4u, v);
  }
}

// ---- CDNA5 async global->LDS stage of one chunk (features + geometry) ----
// GVS addressing: mem = SGPR base + VGPR 32-bit offset. LDS dest offset in VGPR.
__device__ __forceinline__ void lss_issue_chunk(const float* __restrict__ xsrc,
                                                const float* __restrict__ gsrc,
                                                float* sx, float* sg, int tid) {
  unsigned lx = (unsigned)(size_t)sx;      // low 32 bits of generic ptr = LDS offset
  unsigned lg = (unsigned)(size_t)sg;
  // features: LSS_CHUNK*32*4 = 24576 B = 1536 x b128 -> 6 per thread (256 threads)
#pragma unroll
  for (int i = 0; i < (LSS_CHUNK * LSS_C * 4) / (256 * 16); ++i) {
    unsigned off = (unsigned)(tid + i * 256) * 16u;
    asm volatile("global_load_async_to_lds_b128 %0, %1, %2"
                 :: "v"(lx + off), "v"(off), "s"(xsrc) : "memory");
  }
  // geometry: LSS_CHUNK*3*4 = 2304 B = 144 x b128
  if (tid < (LSS_CHUNK * 3 * 4) / 16) {
    unsigned off = (unsigned)tid * 16u;
    asm volatile("global_load_async_to_lds_b128 %0, %1, %2"
                 :: "v"(lg + off), "v"(off), "s"(gsrc) : "memory");
  }
}

// ---- main scatter kernel: double-buffered async LDS pipeline ----
template <int FINAL_LAYOUT>
__global__ void __launch_bounds__(256)
lss_scatter_async(const float* __restrict__ geom, const float* __restrict__ x,
                  float* __restrict__ acc, int nchunks) {
  __shared__ float sx[2][LSS_CHUNK * LSS_C];  // 2 x 24 KB
  __shared__ float sg[2][LSS_CHUNK * 3];      // 2 x 2.25 KB
  const int tid  = (int)threadIdx.x;
  const int wave = tid >> 5;
  const int lane = tid & 31;

  int ck = (int)blockIdx.x;
  if (ck < nchunks)
    lss_issue_chunk(x + (size_t)ck * LSS_CHUNK * LSS_C,
                    geom + (size_t)ck * LSS_CHUNK * 3, sx[0], sg[0], tid);

  int buf = 0;
  for (; ck < nchunks; ck += (int)gridDim.x) {
    asm volatile("s_wait_asynccnt 0x0" ::: "memory");  // my async writes done
    __syncthreads();                                   // everyone's writes visible
    int nxt = ck + (int)gridDim.x;
    if (nxt < nchunks)
      lss_issue_chunk(x + (size_t)nxt * LSS_CHUNK * LSS_C,
                      geom + (size_t)nxt * LSS_CHUNK * 3,
                      sx[buf ^ 1], sg[buf ^ 1], tid);
    // one point per wave per step; lane == channel (conflict-free LDS, 128B atomics)
#pragma unroll 4
    for (int p = wave; p < LSS_CHUNK; p += 8) {
      float g0 = sg[buf][p * 3 + 0];
      float g1 = sg[buf][p * 3 + 1];
      float g2 = sg[buf][p * 3 + 2];
      float v  = sx[buf][p * LSS_C + lane];
      lss_scatter_point<FINAL_LAYOUT>(g0, g1, g2, v, lane, acc);
    }
    buf ^= 1;
  }
}

// ---- tail (non-multiple-of-CHUNK points; unused for this shape but kept correct) ----
template <int FINAL_LAYOUT>
__global__ void lss_scatter_tail(const float* __restrict__ geom,
                                 const float* __restrict__ x,
                                 float* __restrict__ acc, int npts) {
  int t = (int)(blockIdx.x * blockDim.x + threadIdx.x);
  int p = t >> 5, lane = t & 31;
  if (p >= npts) return;
  lss_scatter_point<FINAL_LAYOUT>(geom[p * 3 + 0], geom[p * 3 + 1], geom[p * 3 + 2],
                                  x[(size_t)p * LSS_C + lane], lane, acc);
}

// ---- zero the accumulator (graph-capture safe, vectorized) ----
__global__ void lss_zero(float4* __restrict__ p, int n4) {
  int i = (int)(blockIdx.x * blockDim.x + threadIdx.x);
  if (i < n4) p[i] = make_float4(0.f, 0.f, 0.f, 0.f);
}

// ---- (vox, C) scratch -> (C, X, Y) output, LDS-tiled 32x32 transpose ----
__global__ void __launch_bounds__(256)
lss_transpose(const float* __restrict__ ws, float* __restrict__ out) {
  __shared__ float tile[32][33];             // +1 pad: conflict-free both phases
  const int v0  = (int)blockIdx.x * 32;
  const int tid = (int)threadIdx.x;
#pragma unroll
  for (int k = 0; k < 4; ++k) {
    int e = tid + k * 256;
    int v = e >> 5, c = e & 31;              // consecutive tid -> consecutive addr
    tile[v][c] = ws[(size_t)(v0 + v) * LSS_C + c];
  }
  __syncthreads();
#pragma unroll
  for (int k = 0; k < 4; ++k) {
    int e = tid + k * 256;
    int c = e >> 5, v = e & 31;              // consecutive tid -> consecutive vox
    out[(size_t)c * LSS_VOXN + (size_t)(v0 + v)] = tile[v][c];
  }
}

extern "C" void kernel_launch(void* const* d_in, const int* in_sizes, int n_in,
                              void* d_out, int out_size, void* d_ws, size_t ws_size,
                              hipStream_t stream) {
  const float* geom = (const float*)d_in[0];   // (B,N,D,H,W,3) f32
  const float* x    = (const float*)d_in[1];   // (B,N,D,H,W,32) f32
  float* out        = (float*)d_out;           // (1, 32, 200, 200) f32

  const int npoints = in_sizes[0] / 3;         // 2,770,944
  const bool use_ws = ws_size >= LSS_ACCN * sizeof(float);
  float* acc = use_ws ? (float*)d_ws : out;

  // 1) zero accumulator (5.12 MB)
  {
    int n4 = (int)(LSS_ACCN / 4);              // 320,000 float4
    lss_zero<<<(n4 + 255) / 256, 256, 0, stream>>>((float4*)acc, n4);
  }

  // 2) scatter-add all points (async-LDS double-buffered pipeline)
  const int nchunks = npoints / LSS_CHUNK;
  const int rem     = npoints - nchunks * LSS_CHUNK;
  if (nchunks > 0) {
    int blocks = nchunks < 1024 ? nchunks : 1024;
    if (use_ws)
      lss_scatter_async<0><<<blocks, 256, 0, stream>>>(geom, x, acc, nchunks);
    else
      lss_scatter_async<1><<<blocks, 256, 0, stream>>>(geom, x, acc, nchunks);
  }
  if (rem > 0) {
    int threads = rem * 32;
    int blocks = (threads + 255) / 256;
    const float* gtail = geom + (size_t)nchunks * LSS_CHUNK * 3;
    const float* xtail = x + (size_t)nchunks * LSS_CHUNK * LSS_C;
    if (use_ws)
      lss_scatter_tail<0><<<blocks, 256, 0, stream>>>(gtail, xtail, acc, rem);
    else
      lss_scatter_tail<1><<<blocks, 256, 0, stream>>>(gtail, xtail, acc, rem);
  }

  // 3) layout transform into final (C, X, Y)
  if (use_ws) {
    lss_transpose<<<LSS_VOXN / 32, 256, 0, stream>>>(acc, out);
  }
}